// FiringRateModel_2851858284772
// MI455X (gfx1250) — compile-verified
//
#include <hip/hip_runtime.h>
#include <hip/hip_bf16.h>

typedef __attribute__((ext_vector_type(2))) float v2f;
typedef __attribute__((ext_vector_type(8))) float v8f;

#define T_LEN 262144
#define NTILE (T_LEN / 256)   // 1024 tiles of 256 timesteps

// ---------------------------------------------------------------------------
// Kernel 1: u[t] = Is[t] + sum_{j=0..63} a[j] * Is[t-64+j]   (zeros for t-64+j<0)
// Banded-Toeplitz matmul, K-dim = 80, f32 WMMA 16x16x4:
//   C[m,n] = sum_s A[m,s]*B[s,n],  A[m,s]=a[s-m] (0 outside band),
//   B[s,n] = Is[t0-64 + 16n + s],  u[t0+16n+m] = Is[t0+16n+m] + C[m,n]
// a_pad in LDS makes the A band branch-free; only tile 0 needs B guards.
// ---------------------------------------------------------------------------
__global__ __launch_bounds__(256) void fir_u_wmma(const float* __restrict__ Is,
                                                  const float* __restrict__ a,
                                                  float* __restrict__ u) {
  __shared__ float a_pad[96];   // a_pad[16+d] = a[d] for 0<=d<64, else 0
  {
    const int tid = threadIdx.x;
    if (tid < 96) {
      const int d = tid - 16;
      a_pad[tid] = (d >= 0 && d < 64) ? a[d] : 0.f;
    }
  }
  __syncthreads();

  const int lane = threadIdx.x & 31;
  const int wave = threadIdx.x >> 5;
  const int tile = blockIdx.x * 8 + wave;
  const int t0   = tile * 256;
  const int base = t0 - 64;
  const int m    = lane & 15;   // A row (M); also B column (N)
  const int half = lane >> 4;
  const int kb   = half * 2;    // K pair selector within each 4-wide step

  v8f c = {0.f, 0.f, 0.f, 0.f, 0.f, 0.f, 0.f, 0.f};

  if (tile != 0) {
    // fast path: unconditional loads (wave-uniform branch, EXEC stays full)
#pragma unroll
    for (int p = 0; p < 20; ++p) {
      const int k0 = p * 4 + kb;
      v2f av, bv;
      av.x = a_pad[16 + k0 - m];
      av.y = a_pad[17 + k0 - m];
      bv = *(const v2f*)(Is + (base + 16 * m + k0));   // 8B-aligned
      c = __builtin_amdgcn_wmma_f32_16x16x4_f32(false, av, false, bv,
                                                (short)0, c, false, false);
    }
  } else {
    // tile 0: guard reads below t=0
#pragma unroll
    for (int p = 0; p < 20; ++p) {
      const int k0 = p * 4 + kb;
      v2f av, bv;
      av.x = a_pad[16 + k0 - m];
      av.y = a_pad[17 + k0 - m];
      const int i0 = base + 16 * m + k0;
      bv.x = (i0     >= 0) ? Is[i0]     : 0.f;
      bv.y = (i0 + 1 >= 0) ? Is[i0 + 1] : 0.f;
      c = __builtin_amdgcn_wmma_f32_16x16x4_f32(false, av, false, bv,
                                                (short)0, c, false, false);
    }
  }

#pragma unroll
  for (int r = 0; r < 8; ++r) {
    const int M = r + 8 * half;
    const int t = t0 + 16 * m + M;
    u[t] = Is[t] + c[r];
  }
}

// ---------------------------------------------------------------------------
// Fast wave32 butterfly add: DPP16 for xor 1/2/4/8 (VALU speed), one LDS-pipe
// shuffle for the half-crossing xor16. 0xB1=quad_perm[1,0,3,2] (xor1),
// 0x4E=quad_perm[2,3,0,1] (xor2), 0x141=row_half_mirror (xor7 ~ xor4 stage),
// 0x140=row_mirror (xor15 ~ xor8 stage).
// ---------------------------------------------------------------------------
template <int CTRL>
__device__ __forceinline__ float dpp_xadd(float x) {
  const int y = __builtin_amdgcn_update_dpp(0, __builtin_bit_cast(int, x),
                                            CTRL, 0xf, 0xf, true);
  return x + __builtin_bit_cast(float, y);
}

// ---------------------------------------------------------------------------
// Kernel 2: sequential recurrence, one wave32, latency-optimized.
//   x[i] = u[i] - ( b[63]*f[i-1] + acc[i&63] )     (acc = d>=2 feedback terms)
//   f[i] = softplus( sum_h w2[h]*tanh(w1[h]*x+b1[h]) + b2 )
// Transposed-FIR scatter keeps the 64-tap feedback off the critical path.
// ---------------------------------------------------------------------------
__global__ __launch_bounds__(32) void recurrence(const float* __restrict__ u,
                                                 const float* __restrict__ b,
                                                 const float* __restrict__ w1,
                                                 const float* __restrict__ b1,
                                                 const float* __restrict__ w2,
                                                 const float* __restrict__ b2p,
                                                 float* __restrict__ out) {
  __shared__ float acc[64];
  const int lane = threadIdx.x;  // 0..31, single wave

  acc[lane]      = 0.f;
  acc[lane + 32] = 0.f;

  float w1r2[4], b1r2[4], w2r[4];   // w1,b1 pre-doubled: exp(2y)=exp(2w1*x+2b1)
#pragma unroll
  for (int r = 0; r < 4; ++r) {
    w1r2[r] = 2.0f * w1[lane + 32 * r];
    b1r2[r] = 2.0f * b1[lane + 32 * r];
    w2r[r]  = w2[lane + 32 * r];
  }
  const float cA  = b[62 - lane];
  const float cB  = (lane <= 30) ? b[30 - lane] : 0.f;
  const float b63 = b[63];
  const float gb2 = b2p[0];

  float f_prev = 0.f;
  float accP   = 0.f;
  float uNxt   = u[lane];
  float fOut   = 0.f;

  const int NC = T_LEN / 32;
  for (int ch = 0; ch < NC; ++ch) {
    const float uCur = uNxt;
    const int   nb   = (ch + 1) * 32;
    const int   pidx = (nb + lane < T_LEN) ? (nb + lane) : (T_LEN - 1);
    uNxt = u[pidx];                              // deep prefetch, off chain
    const int i0 = ch * 32;
#pragma unroll 4
    for (int s = 0; s < 32; ++s) {
      const int i    = i0 + s;
      const int slot = (i + 1) & 63;
      const float accNext = acc[slot];   // prefetch d>=2 sum for x[i+1]
      acc[slot] = 0.f;                   // recycle slot

      const float uu = __shfl(uCur, s, 32);           // independent of f chain
      const float x  = uu - fmaf(b63, f_prev, accP);  // d=1 term in registers

      float th[4];
#pragma unroll
      for (int r = 0; r < 4; ++r) {
        const float e = __expf(fmaf(w1r2[r], x, b1r2[r]));  // exp(2y)
        th[r] = 1.0f - 2.0f * __builtin_amdgcn_rcpf(e + 1.0f);
      }
      const float p0   = fmaf(w2r[0], th[0], w2r[1] * th[1]);
      const float p1   = fmaf(w2r[2], th[2], w2r[3] * th[3]);
      float ssum = p0 + p1;

      ssum = dpp_xadd<0xB1>(ssum);    // xor1 (quad_perm)
      ssum = dpp_xadd<0x4E>(ssum);    // xor2 (quad_perm)
      ssum = dpp_xadd<0x141>(ssum);   // row_half_mirror
      ssum = dpp_xadd<0x140>(ssum);   // row_mirror
      ssum += __shfl_xor(ssum, 16, 32);   // cross 16-lane halves

      const float z = ssum + gb2;
      const float f = fmaxf(z, 0.f) + __logf(1.0f + __expf(-fabsf(z)));

      // scatter feedback contributions of f (off the critical path)
      atomicAdd(&acc[(i + 2  + lane) & 63], cA * f);
      atomicAdd(&acc[(i + 34 + lane) & 63], cB * f);

      if (lane == s) fOut = f;
      f_prev = f;
      accP   = accNext;
    }
    out[i0 + lane] = fOut;   // coalesced store once per 32 steps
  }
}

// ---------------------------------------------------------------------------
extern "C" void kernel_launch(void* const* d_in, const int* in_sizes, int n_in,
                              void* d_out, int out_size, void* d_ws, size_t ws_size,
                              hipStream_t stream) {
  const float* Is   = (const float*)d_in[0];
  const float* a    = (const float*)d_in[1];
  const float* b    = (const float*)d_in[2];
  const float* g_w1 = (const float*)d_in[3];
  const float* g_b1 = (const float*)d_in[4];
  const float* g_w2 = (const float*)d_in[5];
  const float* g_b2 = (const float*)d_in[6];
  float* out = (float*)d_out;
  float* u   = (float*)d_ws;   // T floats = 1 MB scratch

  fir_u_wmma<<<NTILE / 8, 256, 0, stream>>>(Is, a, u);
  recurrence<<<1, 32, 0, stream>>>(u, b, g_w1, g_b1, g_w2, g_b2, out);
}